// SphericalToCartesianTransform_19164144075052
// MI455X (gfx1250) — compile-verified
//
#include <hip/hip_runtime.h>
#include <math.h>

// ---------------------------------------------------------------------------
// SphericalToCartesianTransform for MI455X (gfx1250, wave32)
//
// NN search recast as GEMM+bias via V_WMMA_F32_16X16X4_F32:
//   score = theta*(-2 g_theta) + phi*(-2 g_phi) + 1*(g_theta^2+g_phi^2) + 0
// B-operand pack (641 tiles x 256 B = 164 KB) is staged in LDS (160 KB,
// CU-mode safe) so the 9+ GB of per-wave B re-reads hit LDS, not L2.
// ---------------------------------------------------------------------------

typedef __attribute__((ext_vector_type(2))) float v2f;
typedef __attribute__((ext_vector_type(8))) float v8f;

#define DIM 96
#define CENTER 47.5f      // (96-1)/2
#define MAX_SH_TILES 640  // 640 * 32 * 8B = 163840 B = 160 KB LDS

// --- Kernel 1: precompute grid (theta,phi) and pack B operand -------------
// B (4x16 f32) per-lane layout (ISA 7.12.2, mirrors 16x4 A layout):
//   v0: lanes 0-15 -> K=0 row (N=lane), lanes 16-31 -> K=2 row (N=lane-16)
//   v1: lanes 0-15 -> K=1 row,          lanes 16-31 -> K=3 row
__global__ void prep_grid_kernel(const float* __restrict__ gv,
                                 float2* __restrict__ bpack,
                                 int N, int ntiles) {
    int t = blockIdx.x * blockDim.x + threadIdx.x;
    int total = ntiles * 32;
    if (t >= total) return;
    int lane = t & 31;
    int tile = t >> 5;
    int j = tile * 16 + (lane & 15);

    float a = 0.0f, b = 0.0f, c = 1.0e30f;   // pad entries: huge bias
    if (j < N) {
        float gx = gv[j * 3 + 0];
        float gy = gv[j * 3 + 1];
        float gz = gv[j * 3 + 2];
        float gr = sqrtf(gx * gx + gy * gy + gz * gz);
        float gt = acosf(gz / fmaxf(gr, 1e-6f));
        float gp = atan2f(gy, gx);
        a = -2.0f * gt;            // K=0 row
        b = -2.0f * gp;            // K=1 row
        c = gt * gt + gp * gp;     // K=2 row (bias * A's 1.0)
    }
    float2 v;
    if (lane < 16) { v.x = a; v.y = b; }   // K=0 / K=1
    else           { v.x = c; v.y = 0.f; } // K=2 / K=3
    bpack[t] = v;
}

// --- Kernel 2: WMMA nearest-neighbor argmin (LDS-staged B) ----------------
__global__ void __launch_bounds__(256)
nn_wmma_kernel(const float2* __restrict__ bpack,
               int* __restrict__ idx, int ntiles, int shTiles) {
    extern __shared__ float2 sB[];   // shTiles * 32 entries (<= 160 KB)

    // Cooperative stage: global -> LDS, coalesced b64
    {
        int total = shTiles * 32;
        for (int i = threadIdx.x; i < total; i += 256) sB[i] = bpack[i];
    }
    __syncthreads();

    const int lane = threadIdx.x & 31;
    const int tileM = blockIdx.x * (blockDim.x >> 5) + (threadIdx.x >> 5);
    const int baseVox = tileM * 16;

    // Build A operand (selects only; EXEC stays all-ones for WMMA)
    int m = lane & 15;
    int v = baseVox + m;
    int dI = v / (DIM * DIM);
    int rem = v - dI * DIM * DIM;
    int hI = rem / DIM;
    int wI = rem - hI * DIM;
    float cx = (float)dI - CENTER;
    float cy = (float)hI - CENTER;
    float cz = (float)wI - CENTER;
    float rho = sqrtf(cx * cx + cy * cy + cz * cz);
    float theta = acosf(cz / fmaxf(rho, 1e-6f));
    float phi = atan2f(cy, cx);

    v2f A;
    A.x = (lane < 16) ? theta : 1.0f;
    A.y = (lane < 16) ? phi   : 0.0f;

    float minv[8];
    int   mini[8];
#pragma unroll
    for (int r = 0; r < 8; ++r) { minv[r] = 3.0e38f; mini[r] = 0; }

    // Main loop: B tiles from LDS (ds_load_b64 + wmma + argmin update)
    const float2* sp = sB + lane;
    int t = 0;
#pragma unroll 4
    for (; t < shTiles; ++t) {
        float2 bv = sp[t * 32];
        v2f B; B.x = bv.x; B.y = bv.y;
        v8f C = {};
        v8f Dm = __builtin_amdgcn_wmma_f32_16x16x4_f32(
            false, A, false, B, (short)0, C, false, false);
        int col = t * 16 + m;
#pragma unroll
        for (int r = 0; r < 8; ++r) {
            if (Dm[r] < minv[r]) { minv[r] = Dm[r]; mini[r] = col; }
        }
    }
    // Tail tiles (beyond LDS capacity) straight from global/L2
    const float2* gp = bpack + lane;
    for (; t < ntiles; ++t) {
        float2 bv = gp[(size_t)t * 32];
        v2f B; B.x = bv.x; B.y = bv.y;
        v8f C = {};
        v8f Dm = __builtin_amdgcn_wmma_f32_16x16x4_f32(
            false, A, false, B, (short)0, C, false, false);
        int col = t * 16 + m;
#pragma unroll
        for (int r = 0; r < 8; ++r) {
            if (Dm[r] < minv[r]) { minv[r] = Dm[r]; mini[r] = col; }
        }
    }

    // Reduce over the 16 column-residue lanes in each half-wave
#pragma unroll
    for (int r = 0; r < 8; ++r) {
        float mv = minv[r];
        int   mi = mini[r];
#pragma unroll
        for (int off = 8; off >= 1; off >>= 1) {
            float ov = __shfl_xor(mv, off, 16);
            int   oi = __shfl_xor(mi, off, 16);
            if (ov < mv || (ov == mv && oi < mi)) { mv = ov; mi = oi; }
        }
        minv[r] = mv; mini[r] = mi;
    }

    // Lanes 0-15 hold rows 0-7; lanes 16-31 hold rows 8-15
    if (lane == 0) {
#pragma unroll
        for (int r = 0; r < 8; ++r) idx[baseVox + r] = mini[r];
    } else if (lane == 16) {
#pragma unroll
        for (int r = 0; r < 8; ++r) idx[baseVox + 8 + r] = mini[r];
    }
}

// --- Kernel 3: gather displacement + spherical->Cartesian -----------------
__global__ void finalize_kernel(const float* __restrict__ disp,
                                const int* __restrict__ idx,
                                float* __restrict__ out, int V) {
    int v = blockIdx.x * blockDim.x + threadIdx.x;
    if (v >= V) return;
    int dI = v / (DIM * DIM);
    int rem = v - dI * DIM * DIM;
    int hI = rem / DIM;
    int wI = rem - hI * DIM;
    float x = (float)dI - CENTER;
    float y = (float)hI - CENTER;
    float z = (float)wI - CENTER;
    float rho = sqrtf(x * x + y * y + z * z);
    float theta = acosf(z / fmaxf(rho, 1e-6f));
    float phi = atan2f(y, x);

    int j = idx[v];
    float d_rho   = disp[j * 3 + 0];
    float d_theta = disp[j * 3 + 1];
    float d_phi   = disp[j * 3 + 2];

    float st = sinf(theta), ct = cosf(theta);
    float sp = sinf(phi),   cp = cosf(phi);
    float A1 = d_rho;
    float A2 = rho * d_theta;
    float A3 = rho * st * d_phi;

    float ox = A1 * st * cp + A2 * ct * cp - A3 * sp;
    float oy = A1 * st * sp + A2 * ct * sp + A3 * cp;
    float oz = A1 * ct      - A2 * st;

    out[0 * V + v] = ox;
    out[1 * V + v] = oy;
    out[2 * V + v] = oz;
}

extern "C" void kernel_launch(void* const* d_in, const int* in_sizes, int n_in,
                              void* d_out, int out_size, void* d_ws, size_t ws_size,
                              hipStream_t stream) {
    const float* disp = (const float*)d_in[0];   // spherical_displacement (N,3)
    const float* gv   = (const float*)d_in[1];   // grid_vertices (N,3)
    const int N = in_sizes[1] / 3;
    const int ntiles = (N + 15) / 16;            // 641 for N=10242
    const int V = DIM * DIM * DIM;               // 884736

    // workspace layout: [ bpack : ntiles*32 float2 ][ idx : V int ]
    float2* bpack = (float2*)d_ws;
    size_t bpackBytes = (size_t)ntiles * 32 * sizeof(float2);
    size_t off = (bpackBytes + 255) & ~(size_t)255;
    int* idx = (int*)((char*)d_ws + off);

    int prepThreads = ntiles * 32;
    prep_grid_kernel<<<(prepThreads + 255) / 256, 256, 0, stream>>>(
        gv, bpack, N, ntiles);

    const int shTiles = (ntiles < MAX_SH_TILES) ? ntiles : MAX_SH_TILES;
    const size_t shmemBytes = (size_t)shTiles * 32 * sizeof(float2); // <=160KB
    static bool attrSet = false;  // idempotent host-side attribute raise
    if (!attrSet) {
        hipFuncSetAttribute(reinterpret_cast<const void*>(nn_wmma_kernel),
                            hipFuncAttributeMaxDynamicSharedMemorySize,
                            (int)shmemBytes);
        attrSet = true;
    }

    int vtiles = V / 16;                          // 55296 waves
    int blocks = vtiles / 8;                      // 8 waves / 256-thread block
    nn_wmma_kernel<<<blocks, 256, shmemBytes, stream>>>(bpack, idx, ntiles, shTiles);

    finalize_kernel<<<(V + 255) / 256, 256, 0, stream>>>(
        disp, idx, (float*)d_out, V);
}